// PositionAwareLoss_46136538693794
// MI455X (gfx1250) — compile-verified
//
#include <hip/hip_runtime.h>

#define NUM_CLASS 20
#define IGNORE_L  255

constexpr int Dd = 256, Hh = 256, Ww = 32;
constexpr int S   = Dd * Hh * Ww;   // 2,097,152 voxels
constexpr int SD  = Hh * Ww;        // 8192  (D stride)
constexpr int N4  = S / 4;          // 524,288 threads (4 voxels each)
constexpr int BLOCK = 256;
constexpr int NBLK  = N4 / BLOCK;   // 2048 blocks

typedef float f4 __attribute__((ext_vector_type(4)));

// L1 norm of one-hot difference for an edge pair: e(b) - e(a)
__device__ __forceinline__ float g_edge(int a, int b) {
    const int va = (a != IGNORE_L), vb = (b != IGNORE_L);
    if (va & vb) return (a != b) ? 2.0f : 0.0f;
    return (va | vb) ? 1.0f : 0.0f;
}
// L1 norm of 0.5*(e(next) - e(prev)) for interior points
__device__ __forceinline__ float g_int(int a, int b) {
    const int va = (a != IGNORE_L), vb = (b != IGNORE_L);
    if (va & vb) return (a != b) ? 1.0f : 0.0f;
    return (va | vb) ? 0.5f : 0.0f;
}

__global__ __launch_bounds__(BLOCK)
void pal_main(const float* __restrict__ pred, const int* __restrict__ tgt,
              const float* __restrict__ cw, float* __restrict__ part)
{
    __shared__ float s_cw[32];
    __shared__ float s_n[BLOCK / 32];
    __shared__ float s_d[BLOCK / 32];

    const int tid = threadIdx.x;

    // Stage the 20 class weights into LDS via the CDNA5 async DMA path.
    // Wave 0 (lanes 0..31 under wave32) issues per-lane async copies; lanes
    // >= NUM_CLASS clamp to the last element (in-bounds, slots unused).
    if (tid < 32) {
        const int src = (tid < NUM_CLASS) ? tid : (NUM_CLASS - 1);
        const unsigned lds_addr = (unsigned)(unsigned long long)(&s_cw[tid]);
        const unsigned long long gaddr = (unsigned long long)(cw + src);
        asm volatile("global_load_async_to_lds_b32 %0, %1, off"
                     :: "v"(lds_addr), "v"(gaddr) : "memory");
        asm volatile("s_wait_asynccnt 0" ::: "memory");
    }
    __syncthreads();

    const int gtid = blockIdx.x * BLOCK + tid;
    const int v0 = gtid << 2;             // base voxel (flat, W fastest)
    const int w0 = v0 & (Ww - 1);         // 0,4,...,28: 4 voxels share d,h
    const int h  = (v0 >> 5) & (Hh - 1);
    const int d  = v0 >> 13;

    // ---- Issue the streaming prediction loads FIRST (20 x b128, NT hint) ----
    // This is the 168 MB once-read stream that sets the HBM roofline; get all
    // 20 loads in flight before the integer LGA work, and keep them out of
    // L2/WGP$ so the re-read targets array stays resident.
    f4 p[NUM_CLASS];
#pragma unroll
    for (int c = 0; c < NUM_CLASS; ++c)
        p[c] = __builtin_nontemporal_load(
                   reinterpret_cast<const f4*>(pred + (size_t)c * S + v0));

    // ---- Branchless neighbor-target loads (clamped offsets, always in-bounds;
    //      at volume edges the clamped load degenerates to data that the edge
    //      formula does not consume) ----
    const int4 t4 = *reinterpret_cast<const int4*>(tgt + v0);
    int t[4] = { t4.x, t4.y, t4.z, t4.w };

    const int tm = tgt[v0 - (w0 > 0 ? 1 : 0)];
    const int tp = tgt[v0 + (w0 + 4 < Ww ? 4 : 3)];
    const int4 hm = *reinterpret_cast<const int4*>(tgt + v0 - (h > 0      ? Ww : 0));
    const int4 hp = *reinterpret_cast<const int4*>(tgt + v0 + (h < Hh - 1 ? Ww : 0));
    const int4 dm = *reinterpret_cast<const int4*>(tgt + v0 - (d > 0      ? SD : 0));
    const int4 dp = *reinterpret_cast<const int4*>(tgt + v0 + (d < Dd - 1 ? SD : 0));

    const int hmv[4] = { hm.x, hm.y, hm.z, hm.w };
    const int hpv[4] = { hp.x, hp.y, hp.z, hp.w };
    const int dmv[4] = { dm.x, dm.y, dm.z, dm.w };
    const int dpv[4] = { dp.x, dp.y, dp.z, dp.w };

    float wgt[4];
#pragma unroll
    for (int j = 0; j < 4; ++j) {
        const int w    = w0 + j;
        const int prev = (j == 0) ? tm : t[j - 1];
        const int next = (j == 3) ? tp : t[j + 1];
        float g;
        if (w == 0)           g = g_edge(t[j], next);   // first = f[1]-f[0]
        else if (w == Ww - 1) g = g_edge(prev, t[j]);   // last  = f[n-1]-f[n-2]
        else                  g = g_int(prev, next);
        if (h == 0)           g += g_edge(t[j], hpv[j]);
        else if (h == Hh - 1) g += g_edge(hmv[j], t[j]);
        else                  g += g_int(hmv[j], hpv[j]);
        if (d == 0)           g += g_edge(t[j], dpv[j]);
        else if (d == Dd - 1) g += g_edge(dmv[j], t[j]);
        else                  g += g_int(dmv[j], dpv[j]);
        wgt[j] = 1.0f + g;    // ALPHA + BETA * lga
    }

    // ---- Softmax statistics over the 20 classes (per voxel component) ----
    f4 m = p[0];
#pragma unroll
    for (int c = 1; c < NUM_CLASS; ++c) {
        m.x = fmaxf(m.x, p[c].x); m.y = fmaxf(m.y, p[c].y);
        m.z = fmaxf(m.z, p[c].z); m.w = fmaxf(m.w, p[c].w);
    }

    int tc[4];
#pragma unroll
    for (int j = 0; j < 4; ++j) tc[j] = (t[j] == IGNORE_L) ? 0 : t[j];

    f4 sum = (f4)(0.0f);
    float pt[4] = { 0.f, 0.f, 0.f, 0.f };
#pragma unroll
    for (int c = 0; c < NUM_CLASS; ++c) {
        sum.x += __expf(p[c].x - m.x);
        sum.y += __expf(p[c].y - m.y);
        sum.z += __expf(p[c].z - m.z);
        sum.w += __expf(p[c].w - m.w);
        pt[0] = (tc[0] == c) ? p[c].x : pt[0];
        pt[1] = (tc[1] == c) ? p[c].y : pt[1];
        pt[2] = (tc[2] == c) ? p[c].z : pt[2];
        pt[3] = (tc[3] == c) ? p[c].w : pt[3];
    }

    const float lse[4] = { m.x + __logf(sum.x), m.y + __logf(sum.y),
                           m.z + __logf(sum.z), m.w + __logf(sum.w) };

    float num = 0.f, den = 0.f;
#pragma unroll
    for (int j = 0; j < 4; ++j) {
        const bool  valid = (t[j] != IGNORE_L);
        const float wcls  = s_cw[tc[j]];
        const float logp  = pt[j] - lse[j];
        const float loss  = valid ? (-wcls * logp) : 0.0f;
        num += loss * wgt[j];
        den += valid ? wcls : 0.0f;
    }

    // ---- wave32 tree reduction, then cross-wave via LDS ----
#pragma unroll
    for (int off = 16; off > 0; off >>= 1) {
        num += __shfl_down(num, off, 32);
        den += __shfl_down(den, off, 32);
    }
    const int wv = tid >> 5;
    if ((tid & 31) == 0) { s_n[wv] = num; s_d[wv] = den; }
    __syncthreads();
    if (tid == 0) {
        float n = 0.f, dd = 0.f;
#pragma unroll
        for (int i = 0; i < BLOCK / 32; ++i) { n += s_n[i]; dd += s_d[i]; }
        part[blockIdx.x]        = n;
        part[NBLK + blockIdx.x] = dd;
    }
}

// Deterministic second-stage reduction (fixed tree order, no atomics)
__global__ __launch_bounds__(256)
void pal_finalize(const float* __restrict__ part, float* __restrict__ out)
{
    __shared__ float s_n[256];
    __shared__ float s_d[256];
    const int tid = threadIdx.x;
    float n = 0.f, dd = 0.f;
    for (int i = tid; i < NBLK; i += 256) { n += part[i]; dd += part[NBLK + i]; }
    s_n[tid] = n; s_d[tid] = dd;
    __syncthreads();
    for (int off = 128; off > 0; off >>= 1) {
        if (tid < off) { s_n[tid] += s_n[tid + off]; s_d[tid] += s_d[tid + off]; }
        __syncthreads();
    }
    if (tid == 0) out[0] = s_n[0] / s_d[0];
}

extern "C" void kernel_launch(void* const* d_in, const int* in_sizes, int n_in,
                              void* d_out, int out_size, void* d_ws, size_t ws_size,
                              hipStream_t stream)
{
    const float* pred = (const float*)d_in[0];  // [1,20,256,256,32] f32
    const int*   tgt  = (const int*)d_in[1];    // [1,256,256,32]    i32
    const float* cw   = (const float*)d_in[2];  // [20]              f32
    float* part = (float*)d_ws;                 // 2 * NBLK floats (16 KB)
    float* out  = (float*)d_out;                // scalar f32

    pal_main<<<NBLK, BLOCK, 0, stream>>>(pred, tgt, cw, part);
    pal_finalize<<<1, 256, 0, stream>>>(part, out);
}